// GAT4_16896401342686
// MI455X (gfx1250) — compile-verified
//
#include <hip/hip_runtime.h>
#include <hip/hip_bf16.h>

typedef float v2f __attribute__((ext_vector_type(2)));
typedef float v8f __attribute__((ext_vector_type(8)));

#define NEG_ATT 0.2f
#define NEG_ACT 0.01f

// ---------------------------------------------------------------------------
// WMMA f32 GEMM: H[N,Co] = X[N,Ci] @ W[Ci,Co].
// One wave32 computes one 16x16 output tile; K stepped by 4 via
// V_WMMA_F32_16X16X4_F32 (full f32 precision, matches the f32 reference).
// Lane layouts per CDNA5 ISA 7.12.2 (32-bit A 16x4, 32-bit C/D 16x16).
// ---------------------------------------------------------------------------
__global__ __launch_bounds__(128) void gat_gemm_wmma_f32(
    const float* __restrict__ X, const float* __restrict__ W,
    float* __restrict__ H, int N, int Ci, int Co)
{
    const int lane   = threadIdx.x & 31;
    const int wave   = threadIdx.x >> 5;
    const int tilesN = Co >> 4;
    const int tilesT = (N >> 4) * tilesN;
    const int tile   = blockIdx.x * 4 + wave;
    if (tile >= tilesT) return;           // wave-uniform: EXEC stays all-ones

    const int tm   = tile / tilesN;
    const int tn   = tile - tm * tilesN;
    const int rc   = lane & 15;           // A-row / B-col / D-col
    const int half = lane >> 4;           // 0: K pair base 0, 1: base 2
    const int m0   = tm << 4;
    const int n0   = tn << 4;

    v8f acc = {};
    const float* xrow = X + (size_t)(m0 + rc) * Ci;
    for (int k0 = 0; k0 < Ci; k0 += 4) {
        // A 16x4: VGPR v holds K = k0 + v + 2*half  -> contiguous pair
        v2f a = *(const v2f*)(xrow + k0 + 2 * half);
        // B 4x16: lane holds column rc; VGPR v holds K = k0 + v + 2*half
        v2f b;
        b.x = W[(size_t)(k0 + 2 * half + 0) * Co + n0 + rc];
        b.y = W[(size_t)(k0 + 2 * half + 1) * Co + n0 + rc];
        acc = __builtin_amdgcn_wmma_f32_16x16x4_f32(
            /*neg_a=*/false, a, /*neg_b=*/false, b,
            /*c_mod=*/(short)0, acc, /*reuse_a=*/false, /*reuse_b=*/false);
    }
    // D 16x16: VGPR r -> row (r + 8*half), col rc
    float* hb = H + (size_t)m0 * Co + n0 + rc;
#pragma unroll
    for (int r = 0; r < 8; ++r)
        hb[(size_t)(r + 8 * half) * Co] = acc[r];
}

// ---------------------------------------------------------------------------
// One-time: expand int64 edge_index [2,E] into int32 src/dst with the N
// self-loop edges appended (indices are reused by all 4 layers from L2).
// ---------------------------------------------------------------------------
__global__ void gat_build_edges(const long long* __restrict__ ei,
                                int* __restrict__ S32, int* __restrict__ D32,
                                int E, int N)
{
    int i = blockIdx.x * blockDim.x + threadIdx.x;
    if (i >= E + N) return;
    if (i < E) { S32[i] = (int)ei[i]; D32[i] = (int)ei[(size_t)E + i]; }
    else       { S32[i] = D32[i] = i - E; }
}

// ---------------------------------------------------------------------------
// Per-node attention logit halves: As[n] = h[n].a_src, Ad[n] = h[n].a_dst
// One wave per node, wave32 shuffle reduction.
// ---------------------------------------------------------------------------
__global__ __launch_bounds__(128) void gat_node_alphas(
    const float* __restrict__ H, const float* __restrict__ a_s,
    const float* __restrict__ a_d, float* __restrict__ As,
    float* __restrict__ Ad, int N, int Co)
{
    const int lane = threadIdx.x & 31;
    const int node = blockIdx.x * 4 + (threadIdx.x >> 5);
    if (node >= N) return;
    const float* hrow = H + (size_t)node * Co;
    float ss = 0.f, sd = 0.f;
    for (int c = lane; c < Co; c += 32) {
        float v = hrow[c];
        ss += v * a_s[c];
        sd += v * a_d[c];
    }
#pragma unroll
    for (int off = 16; off > 0; off >>= 1) {
        ss += __shfl_down(ss, off, 32);
        sd += __shfl_down(sd, off, 32);
    }
    if (lane == 0) { As[node] = ss; Ad[node] = sd; }
}

// ---------------------------------------------------------------------------
// Order-preserving float <-> uint mapping for atomicMax-based segment max.
// ---------------------------------------------------------------------------
__device__ __forceinline__ unsigned fenc(float f) {
    unsigned u = __float_as_uint(f);
    return (u & 0x80000000u) ? ~u : (u | 0x80000000u);
}
__device__ __forceinline__ float fdec(unsigned u) {
    return __uint_as_float((u & 0x80000000u) ? (u ^ 0x80000000u) : ~u);
}

// Init per-layer state: segment max (encoded -huge), segment sum = 0,
// Out[n,c] = bias[c] (aggregation accumulates on top of bias).
__global__ void gat_init_layer(unsigned* __restrict__ Menc, float* __restrict__ S,
                               float* __restrict__ Out, const float* __restrict__ bias,
                               int N, int Co)
{
    size_t i = (size_t)blockIdx.x * blockDim.x + threadIdx.x;
    if (i < (size_t)N) { Menc[i] = 0u; S[i] = 0.f; }
    size_t total = (size_t)N * (size_t)Co;
    if (i < total) Out[i] = bias[i & (size_t)(Co - 1)];
}

// Pass 1: e = leakyrelu(As[src]+Ad[dst], 0.2); segment max via atomicMax.
__global__ void gat_edge_logits_max(const int* __restrict__ SRC,
                                    const int* __restrict__ DST,
                                    const float* __restrict__ As,
                                    const float* __restrict__ Ad,
                                    float* __restrict__ Ebuf,
                                    unsigned* __restrict__ Menc, int EE)
{
    int i = blockIdx.x * blockDim.x + threadIdx.x;
    if (i >= EE) return;
    int s = SRC[i], d = DST[i];
    float e = As[s] + Ad[d];
    e = (e > 0.f) ? e : NEG_ATT * e;
    Ebuf[i] = e;
    atomicMax(&Menc[d], fenc(e));
}

// Pass 2: ex = exp(e - m[dst]); segment sum via atomicAdd.
__global__ void gat_edge_exp_sum(const int* __restrict__ DST,
                                 const unsigned* __restrict__ Menc,
                                 float* __restrict__ Ebuf,
                                 float* __restrict__ S, int EE)
{
    int i = blockIdx.x * blockDim.x + threadIdx.x;
    if (i >= EE) return;
    int d = DST[i];
    float ex = __expf(Ebuf[i] - fdec(Menc[d]));
    Ebuf[i] = ex;
    atomicAdd(&S[d], ex);
}

// Pass 3: alpha = ex / s[dst]  (precompute once per edge).
__global__ void gat_edge_norm(const int* __restrict__ DST,
                              const float* __restrict__ S,
                              float* __restrict__ Ebuf, int EE)
{
    int i = blockIdx.x * blockDim.x + threadIdx.x;
    if (i >= EE) return;
    Ebuf[i] = Ebuf[i] / S[DST[i]];
}

// Pass 4: Out[dst] += alpha * H[src]. One thread per (edge, channel);
// a wave covers 32 consecutive channels of one edge -> coalesced H reads
// and coalesced L2 atomics; SRC/DST/Alpha reads are wave-uniform.
__global__ __launch_bounds__(256) void gat_edge_aggregate(
    const int* __restrict__ SRC, const int* __restrict__ DST,
    const float* __restrict__ H, const float* __restrict__ Alpha,
    float* __restrict__ Out, int EE, int Co, int logCo)
{
    size_t tid = (size_t)blockIdx.x * blockDim.x + threadIdx.x;
    size_t total = (size_t)EE * (size_t)Co;
    if (tid >= total) return;
    int i = (int)(tid >> logCo);
    int c = (int)(tid & (size_t)(Co - 1));
    int s = SRC[i], d = DST[i];
    atomicAdd(&Out[(size_t)d * Co + c], H[(size_t)s * Co + c] * Alpha[i]);
}

// Inter-layer activation: LeakyReLU(0.01), in place.
__global__ void gat_leaky_act(float* __restrict__ X, size_t n)
{
    size_t i = (size_t)blockIdx.x * blockDim.x + threadIdx.x;
    if (i < n) { float v = X[i]; X[i] = (v > 0.f) ? v : NEG_ACT * v; }
}

// ---------------------------------------------------------------------------
static void run_gat_layer(const float* in, const float* Wm, const float* a_s,
                          const float* a_d, const float* bias,
                          float* Hbuf, float* Out,
                          const int* SRC, const int* DST,
                          float* As, float* Ad, unsigned* Menc, float* S,
                          float* Ebuf, int N, int EE, int Ci, int Co, int logCo,
                          bool act, hipStream_t stream)
{
    const int tiles = (N >> 4) * (Co >> 4);
    gat_gemm_wmma_f32<<<(tiles + 3) / 4, 128, 0, stream>>>(in, Wm, Hbuf, N, Ci, Co);
    gat_node_alphas<<<(N + 3) / 4, 128, 0, stream>>>(Hbuf, a_s, a_d, As, Ad, N, Co);
    size_t initN = (size_t)N * Co;
    gat_init_layer<<<(unsigned)((initN + 255) / 256), 256, 0, stream>>>(Menc, S, Out, bias, N, Co);
    gat_edge_logits_max<<<(EE + 255) / 256, 256, 0, stream>>>(SRC, DST, As, Ad, Ebuf, Menc, EE);
    gat_edge_exp_sum<<<(EE + 255) / 256, 256, 0, stream>>>(DST, Menc, Ebuf, S, EE);
    gat_edge_norm<<<(EE + 255) / 256, 256, 0, stream>>>(DST, S, Ebuf, EE);
    size_t aggN = (size_t)EE * Co;
    gat_edge_aggregate<<<(unsigned)((aggN + 255) / 256), 256, 0, stream>>>(
        SRC, DST, Hbuf, Ebuf, Out, EE, Co, logCo);
    if (act) {
        gat_leaky_act<<<(unsigned)((initN + 255) / 256), 256, 0, stream>>>(Out, initN);
    }
}

extern "C" void kernel_launch(void* const* d_in, const int* in_sizes, int n_in,
                              void* d_out, int out_size, void* d_ws, size_t ws_size,
                              hipStream_t stream)
{
    // Inputs (setup_inputs order): x, edge_index, then {W,a_s,a_d,b} x4
    const float*     x  = (const float*)d_in[0];
    const long long* ei = (const long long*)d_in[1];   // int64 [2, E]
    const int Cin = 64, H1 = 128, H2 = 64, Cout = 32;
    const int N  = in_sizes[0] / Cin;
    const int E  = in_sizes[1] / 2;
    const int EE = E + N;                              // + self loops

    const float* W1 = (const float*)d_in[2];
    const float* a1s = (const float*)d_in[3];
    const float* a1d = (const float*)d_in[4];
    const float* b1 = (const float*)d_in[5];
    const float* W2 = (const float*)d_in[6];
    const float* a2s = (const float*)d_in[7];
    const float* a2d = (const float*)d_in[8];
    const float* b2 = (const float*)d_in[9];
    const float* W3 = (const float*)d_in[10];
    const float* a3s = (const float*)d_in[11];
    const float* a3d = (const float*)d_in[12];
    const float* b3 = (const float*)d_in[13];
    const float* W4 = (const float*)d_in[14];
    const float* a4s = (const float*)d_in[15];
    const float* a4d = (const float*)d_in[16];
    const float* b4 = (const float*)d_in[17];

    // Workspace carve-up (~62 MB)
    float*    B0   = (float*)d_ws;                 // GEMM output h  [N,128]
    float*    B1   = B0 + (size_t)N * 128;         // feature buffer [N,128]
    float*    As   = B1 + (size_t)N * 128;         // [N]
    float*    Ad   = As + N;                       // [N]
    unsigned* Menc = (unsigned*)(Ad + N);          // [N]
    float*    S    = (float*)(Menc + N);           // [N]
    float*    Ebuf = S + N;                        // [EE]
    int*      SRC  = (int*)(Ebuf + EE);            // [EE]
    int*      DST  = SRC + EE;                     // [EE]

    float* out = (float*)d_out;                    // [N, 32] f32

    // Materialize int32 indices (+self loops) once; reused by all layers.
    gat_build_edges<<<(EE + 255) / 256, 256, 0, stream>>>(ei, SRC, DST, E, N);

    // Layer 1: x(64) -> B1(128)
    run_gat_layer(x,  W1, a1s, a1d, b1, B0, B1, SRC, DST, As, Ad, Menc, S, Ebuf,
                  N, EE, Cin, H1, 7, /*act=*/true, stream);
    // Layer 2: B1(128) -> B1(64)   (GEMM finishes before Out init overwrites B1)
    run_gat_layer(B1, W2, a2s, a2d, b2, B0, B1, SRC, DST, As, Ad, Menc, S, Ebuf,
                  N, EE, H1, H2, 6, /*act=*/true, stream);
    // Layer 3: B1(64) -> B1(128)
    run_gat_layer(B1, W3, a3s, a3d, b3, B0, B1, SRC, DST, As, Ad, Menc, S, Ebuf,
                  N, EE, H2, H1, 7, /*act=*/true, stream);
    // Layer 4: B1(128) -> d_out(32), no activation
    run_gat_layer(B1, W4, a4s, a4d, b4, B0, out, SRC, DST, As, Ad, Menc, S, Ebuf,
                  N, EE, H1, Cout, 5, /*act=*/false, stream);
    (void)n_in; (void)out_size; (void)ws_size;
}